// RezeroGCN_27247272526303
// MI455X (gfx1250) — compile-verified
//
#include <hip/hip_runtime.h>
#include <hip/hip_bf16.h>

typedef __attribute__((ext_vector_type(2))) float v2f;
typedef __attribute__((ext_vector_type(8))) float v8f;

#define HID 128
#define OUTC 64
#define NLAYERS 4

// ---------------------------------------------------------------- utilities

__global__ void fill_kernel(float* __restrict__ p, float val, long n) {
    long i = (long)blockIdx.x * blockDim.x + threadIdx.x;
    if (i < n) p[i] = val;
}

// Native CDNA5 fp32 global atomic add, no-return form (STOREcnt-tracked).
// Inline asm guarantees GLOBAL_ATOMIC_ADD_F32 rather than a CAS loop.
__device__ __forceinline__ void atomic_add_f32(float* p, float v) {
    asm volatile("global_atomic_add_f32 %0, %1, off"
                 :
                 : "v"(p), "v"(v)
                 : "memory");
}

// deg[dst] += 1 over all real edges (deg pre-initialized to 1.0 for self-loop)
__global__ void degree_kernel(const long long* __restrict__ dst, float* __restrict__ deg,
                              long n_edges) {
    long e = (long)blockIdx.x * blockDim.x + threadIdx.x;
    if (e < n_edges) atomic_add_f32(deg + dst[e], 1.0f);
}

// in-place deg -> deg^{-1/2}
__global__ void dinv_kernel(float* __restrict__ deg, long n_nodes) {
    long v = (long)blockIdx.x * blockDim.x + threadIdx.x;
    if (v < n_nodes) {
        float d = deg[v];
        deg[v] = (d > 0.0f) ? rsqrtf(d) : 0.0f;
    }
}

// ------------------------------------------------------- WMMA fp32 GEMM
// C[n_rows x NCOL] = A[n_rows x 128] * W[128 x NCOL] (+ bias)
// One block computes a 16-row M tile; wave w owns N-columns [16w, 16w+16).
// blockDim.x = 32 * (NCOL/16).
template <int NCOL>
__global__ void gemm_wmma_kernel(const float* __restrict__ A,
                                 const float* __restrict__ W,
                                 const float* __restrict__ bias,
                                 float* __restrict__ C, long n_rows) {
    constexpr int LDA = HID + 4;                 // padded LDS stride (dodge bank conflicts)
    __shared__ float As[16 * LDA];

    const int tid   = threadIdx.x;
    const int wave  = tid >> 5;
    const int lane  = tid & 31;
    const long m0   = (long)blockIdx.x * 16;

    // stage 16x128 A tile into LDS, coalesced float4 loads
    const int nthr = blockDim.x;
    for (int i = tid; i < (16 * HID) / 4; i += nthr) {
        int flat = i * 4;
        int r = flat >> 7;        // /128
        int c = flat & 127;
        float4 vv = make_float4(0.f, 0.f, 0.f, 0.f);
        long row = m0 + r;
        if (row < n_rows) vv = *(const float4*)(A + row * HID + c);
        *(float4*)(&As[r * LDA + c]) = vv;
    }
    __syncthreads();

    const int n0    = wave * 16;
    const int sub   = lane & 15;   // A: row-in-tile ; B/D: column-in-tile
    const int khalf = lane >> 4;   // selects K pair {0,1} vs {2,3} of each step

    v8f acc = {};
    #pragma unroll
    for (int kb = 0; kb < HID; kb += 4) {
        // A fragment (16x4): lane holds A[sub][kb + 2*khalf + {0,1}]
        v2f a = *(const v2f*)(&As[sub * LDA + kb + 2 * khalf]);
        // B fragment (4x16): lane holds W[kb + 2*khalf + {0,1}][n0 + sub]
        const float* wp = W + (long)(kb + 2 * khalf) * NCOL + n0 + sub;
        v2f b;
        b.x = wp[0];
        b.y = wp[NCOL];
        acc = __builtin_amdgcn_wmma_f32_16x16x4_f32(
            /*neg_a=*/false, a, /*neg_b=*/false, b,
            /*c_mod=*/(short)0, acc, /*reuse_a=*/false, /*reuse_b=*/false);
    }

    // D layout: VGPR r -> row (r + 8*khalf), col = sub
    const int col = n0 + sub;
    float bv = (bias != nullptr) ? bias[col] : 0.0f;
    #pragma unroll
    for (int r = 0; r < 8; ++r) {
        long row = m0 + r + khalf * 8;
        if (row < n_rows) C[row * NCOL + col] = acc[r] + bv;
    }
}

// ----------------------------------------------------- edge scatter-add
// One wave per edge; lane l moves channels [4l, 4l+4): agg[dst] += h[src]*norm
__global__ void scatter_kernel(const long long* __restrict__ src,
                               const long long* __restrict__ dst,
                               const float* __restrict__ dinv,
                               const float* __restrict__ h,
                               float* __restrict__ agg, long n_edges) {
    long gid  = (long)blockIdx.x * blockDim.x + threadIdx.x;
    long e    = gid >> 5;
    int  lane = (int)(gid & 31);
    if (e >= n_edges) return;
    long s = src[e];
    long d = dst[e];
    float w = dinv[s] * dinv[d];
    float4 vv = *(const float4*)(h + s * HID + lane * 4);
    float* ap = agg + d * HID + lane * 4;
    atomic_add_f32(ap + 0, vv.x * w);
    atomic_add_f32(ap + 1, vv.y * w);
    atomic_add_f32(ap + 2, vv.z * w);
    atomic_add_f32(ap + 3, vv.w * w);
}

// ------------------------------------------- fused self-loop + bias + tanh + rezero
// h_io holds hW on entry; on exit holds the new layer input.
__global__ void finalize_kernel(const float* __restrict__ inp,
                                float* __restrict__ h_io,
                                const float* __restrict__ agg,
                                const float* __restrict__ dinv,
                                const float* __restrict__ b,
                                const float* __restrict__ alpha,
                                const float* __restrict__ beta,
                                int layer, long n_nodes) {
    long i = (long)blockIdx.x * blockDim.x + threadIdx.x;
    long total = n_nodes * HID;
    if (i >= total) return;
    long v = i >> 7;       // /128
    int  c = (int)(i & 127);
    float di = dinv[v];
    float val = agg[i] + h_io[i] * di * di + b[c];   // self-loop edge + bias
    float t = tanhf(val);
    h_io[i] = beta[layer] * inp[i] + alpha[layer] * t;
}

// ------------------------------------------------ in-place log-softmax (64 wide)
__global__ void log_softmax64_kernel(float* __restrict__ logits, long n_nodes) {
    long gid = (long)blockIdx.x * blockDim.x + threadIdx.x;
    long v   = gid >> 5;
    int lane = (int)(gid & 31);
    if (v >= n_nodes) return;
    float* row = logits + v * OUTC;
    float v0 = row[lane];
    float v1 = row[lane + 32];
    float m = fmaxf(v0, v1);
    #pragma unroll
    for (int off = 16; off > 0; off >>= 1) m = fmaxf(m, __shfl_xor(m, off, 32));
    float s = expf(v0 - m) + expf(v1 - m);
    #pragma unroll
    for (int off = 16; off > 0; off >>= 1) s += __shfl_xor(s, off, 32);
    float l = logf(s);
    row[lane]      = v0 - m - l;
    row[lane + 32] = v1 - m - l;
}

// ---------------------------------------------------------------- launcher

extern "C" void kernel_launch(void* const* d_in, const int* in_sizes, int n_in,
                              void* d_out, int out_size, void* d_ws, size_t ws_size,
                              hipStream_t stream) {
    const float*     x      = (const float*)d_in[0];
    const long long* ei     = (const long long*)d_in[1];   // [2, E] int64
    const float*     W_in   = (const float*)d_in[2];
    const float*     b_in   = (const float*)d_in[3];
    const float*     W_conv = (const float*)d_in[4];       // [4,128,128]
    const float*     b_conv = (const float*)d_in[5];       // [4,128]
    const float*     alpha  = (const float*)d_in[6];
    const float*     beta   = (const float*)d_in[7];
    const float*     W_out  = (const float*)d_in[8];       // [128,64]
    const float*     b_out  = (const float*)d_in[9];
    float*           out    = (float*)d_out;

    const long n_nodes = (long)in_sizes[0] / HID;
    const long n_edges = (long)in_sizes[1] / 2;
    const long long* e_src = ei;
    const long long* e_dst = ei + n_edges;

    // workspace layout
    float* dinv = (float*)d_ws;                 // n_nodes   (deg -> dinv in place)
    float* bufA = dinv + n_nodes;               // n_nodes*128
    float* bufB = bufA + n_nodes * HID;         // n_nodes*128
    float* agg  = bufB + n_nodes * HID;         // n_nodes*128

    const int T = 256;
    const long featN = n_nodes * HID;

    // 1) degree (self-loop -> init 1.0), then deg^-1/2 in place
    fill_kernel<<<(unsigned)((n_nodes + T - 1) / T), T, 0, stream>>>(dinv, 1.0f, n_nodes);
    degree_kernel<<<(unsigned)((n_edges + T - 1) / T), T, 0, stream>>>(e_dst, dinv, n_edges);
    dinv_kernel<<<(unsigned)((n_nodes + T - 1) / T), T, 0, stream>>>(dinv, n_nodes);

    const unsigned gemm_blocks = (unsigned)((n_nodes + 15) / 16);

    // 2) input linear: bufA = x @ W_in + b_in
    gemm_wmma_kernel<HID><<<gemm_blocks, 8 * 32, 0, stream>>>(x, W_in, b_in, bufA, n_nodes);

    // 3) conv layers
    float* inp = bufA;
    float* hw  = bufB;
    for (int i = 0; i < NLAYERS; ++i) {
        gemm_wmma_kernel<HID><<<gemm_blocks, 8 * 32, 0, stream>>>(
            inp, W_conv + (long)i * HID * HID, nullptr, hw, n_nodes);
        fill_kernel<<<(unsigned)((featN + T - 1) / T), T, 0, stream>>>(agg, 0.0f, featN);
        long scatter_threads = n_edges * 32;
        scatter_kernel<<<(unsigned)((scatter_threads + T - 1) / T), T, 0, stream>>>(
            e_src, e_dst, dinv, hw, agg, n_edges);
        finalize_kernel<<<(unsigned)((featN + T - 1) / T), T, 0, stream>>>(
            inp, hw, agg, dinv, b_conv + (long)i * HID, alpha, beta, i, n_nodes);
        float* t = inp; inp = hw; hw = t;   // inp now holds the new input
    }

    // 4) output linear into d_out, then in-place log-softmax
    gemm_wmma_kernel<OUTC><<<gemm_blocks, 4 * 32, 0, stream>>>(inp, W_out, b_out, out, n_nodes);
    long sm_threads = n_nodes * 32;
    log_softmax64_kernel<<<(unsigned)((sm_threads + T - 1) / T), T, 0, stream>>>(out, n_nodes);
}